// InceptionGNNStack_75161927680555
// MI455X (gfx1250) — compile-verified
//
#include <hip/hip_runtime.h>
#include <hip/hip_bf16.h>

// ---------------------------------------------------------------------------
// InceptionGNNStack (3-layer, 3-path, 4-head GATv2) for MI455X / gfx1250.
//
// Dense transforms use V_WMMA_F32_16X16X4_F32 (fp32-exact matrix pipe).
// Edge phase: 3 atomic passes implementing segment softmax exactly as the
// reference (max -> exp/sum -> normalize/scatter). Epilogue fuses bias,
// ReLU and LayerNorm with wave32 shuffle reductions.
// ---------------------------------------------------------------------------

typedef __attribute__((ext_vector_type(2))) float v2f;
typedef __attribute__((ext_vector_type(8))) float v8f;

#define DD   96      // feature dim
#define PP   3       // paths
#define HH   4       // heads
#define CC   8       // channels per head
#define PH   12      // P*H
#define NEG_SLOPE 0.2f
#define LN_EPS    1e-5f
#define LDS_PITCH 97 // 16xDD tile pitch (odd -> conflict-free across M)

// -------------------------- helpers ----------------------------------------

__device__ __forceinline__ void atomicMaxF32(float* addr, float v) {
  // Monotone float ordering trick: non-negative floats order like ints,
  // negative floats order inversely like uints. Init value must be -inf.
  if (v >= 0.0f) {
    atomicMax((int*)addr, __float_as_int(v));
  } else {
    atomicMin((unsigned int*)addr, __float_as_uint(v));
  }
}

// -------------------------- init -------------------------------------------

// One thread per (node, p*h); also zeroes the 8 accumulator channels.
__global__ void init_kernel(float* __restrict__ maxb, float* __restrict__ sumb,
                            float* __restrict__ ybuf, int n12) {
  int i = blockIdx.x * blockDim.x + threadIdx.x;
  if (i < n12) {
    maxb[i] = -INFINITY;
    sumb[i] = 0.0f;
#pragma unroll
    for (int j = 0; j < 8; ++j) ybuf[(size_t)i * 8 + j] = 0.0f;
  }
}

// -------------------------- WMMA GEMM --------------------------------------
// Computes XL[i, p*32+j] = sum_d h[i,d]*Wl[l,p,d,j] + bl[l,p,j]   (and XR).
// Block = 128 threads (4 waves), one 16-row strip of h per block staged in
// LDS; 12 output tiles (6 col-tiles x {L,R}) split 3 per wave.
__global__ void gemm_xlxr_kernel(const float* __restrict__ h,
                                 const float* __restrict__ Wl,
                                 const float* __restrict__ Wr,
                                 const float* __restrict__ bl,
                                 const float* __restrict__ br,
                                 float* __restrict__ XL,
                                 float* __restrict__ XR,
                                 int layer, int Nn) {
  __shared__ float sA[16 * LDS_PITCH];

  const int row0 = blockIdx.x * 16;
  const int tid  = threadIdx.x;

  // Cooperative coalesced load of the 16 x 96 activation tile (zero-padded).
  for (int i = tid; i < 16 * DD; i += 128) {
    int r = i / DD, c = i - r * DD;
    int gr = row0 + r;
    sA[r * LDS_PITCH + c] = (gr < Nn) ? h[(size_t)gr * DD + c] : 0.0f;
  }
  __syncthreads();

  const int wave  = tid >> 5;
  const int lane  = tid & 31;
  const int l16   = lane & 15;        // M for A-frag, N for B/C-frags
  const int kpair = (lane >> 4) * 2;  // K sub-pair selected by lane half
  const int mhalf = (lane >> 4) * 8;  // C/D row offset for upper lane half

#pragma unroll
  for (int it = 0; it < 3; ++it) {
    const int t      = wave * 3 + it;     // 0..11
    const int side   = (t >= 6);          // 0 -> L, 1 -> R
    const int ct     = t % 6;             // column tile 0..5
    const int p      = ct >> 1;
    const int subcol = (ct & 1) * 16;

    const float* W    = (side ? Wr : Wl) + ((size_t)layer * PP + p) * DD * 32;
    const float* bvec = (side ? br : bl) + ((size_t)layer * PP + p) * 32;

    v8f acc = {};
    // A 16x4 frag: lane holds A[M=l16][k0+kpair+{0,1}]
    // B 4x16 frag: lane holds B[K=k0+kpair+{0,1}][N=l16]
    for (int k0 = 0; k0 < DD; k0 += 4) {
      v2f a, b;
      a.x = sA[l16 * LDS_PITCH + k0 + kpair + 0];
      a.y = sA[l16 * LDS_PITCH + k0 + kpair + 1];
      b.x = W[(size_t)(k0 + kpair + 0) * 32 + subcol + l16];
      b.y = W[(size_t)(k0 + kpair + 1) * 32 + subcol + l16];
      acc = __builtin_amdgcn_wmma_f32_16x16x4_f32(
          /*neg_a=*/false, a, /*neg_b=*/false, b,
          /*c_mod=*/(short)0, acc, /*reuse_a=*/false, /*reuse_b=*/false);
    }

    float* OUT      = side ? XR : XL;
    const int col   = p * 32 + subcol + l16;   // column in [0,96)
    const float bv  = bvec[subcol + l16];
#pragma unroll
    for (int r = 0; r < 8; ++r) {
      int gr = row0 + mhalf + r;               // C layout: VGPR r -> M=r(+8)
      if (gr < Nn) OUT[(size_t)gr * DD + col] = acc[r] + bv;
    }
  }
}

// -------------------------- pass A: logits + segment max --------------------
__global__ void logits_kernel(const float* __restrict__ XL,
                              const float* __restrict__ XR,
                              const int* __restrict__ ei,
                              const float* __restrict__ att,
                              float* __restrict__ ELOG,
                              float* __restrict__ MAXB,
                              int layer, int Ep, int E) {
  int e = blockIdx.x * blockDim.x + threadIdx.x;
  if (e >= Ep) return;
  int src, dst;
  if (e < E) { src = ei[e]; dst = ei[E + e]; }
  else       { src = dst = e - E; }   // self loops appended after real edges

  const float4* xlv  = (const float4*)(XL + (size_t)src * DD);
  const float4* xrv  = (const float4*)(XR + (size_t)dst * DD);
  const float*  attl = att + (size_t)layer * DD;   // [P,H,C] flat = 96

#pragma unroll
  for (int ph = 0; ph < PH; ++ph) {                // channel base = ph*8
    float s = 0.0f;
#pragma unroll
    for (int q = 0; q < 2; ++q) {
      float4 a = xlv[ph * 2 + q];
      float4 b = xrv[ph * 2 + q];
      const float* av = attl + ph * 8 + q * 4;
      float m;
      m = a.x + b.x; s += (m > 0.0f ? m : NEG_SLOPE * m) * av[0];
      m = a.y + b.y; s += (m > 0.0f ? m : NEG_SLOPE * m) * av[1];
      m = a.z + b.z; s += (m > 0.0f ? m : NEG_SLOPE * m) * av[2];
      m = a.w + b.w; s += (m > 0.0f ? m : NEG_SLOPE * m) * av[3];
    }
    ELOG[(size_t)e * PH + ph] = s;
    atomicMaxF32(&MAXB[(size_t)dst * PH + ph], s);
  }
}

// -------------------------- pass B: exp + segment sum -----------------------
__global__ void expsum_kernel(float* __restrict__ ELOG,
                              const float* __restrict__ MAXB,
                              float* __restrict__ SUMB,
                              const int* __restrict__ ei, int Ep, int E) {
  int e = blockIdx.x * blockDim.x + threadIdx.x;
  if (e >= Ep) return;
  int dst = (e < E) ? ei[E + e] : (e - E);
#pragma unroll
  for (int ph = 0; ph < PH; ++ph) {
    float mx = MAXB[(size_t)dst * PH + ph];
    float ex = __expf(ELOG[(size_t)e * PH + ph] - mx);
    ELOG[(size_t)e * PH + ph] = ex;
    atomicAdd(&SUMB[(size_t)dst * PH + ph], ex);
  }
}

// -------------------------- pass C: weighted scatter ------------------------
// grid.y = path; one thread per edge handles 4 heads x 8 channels.
__global__ void scatter_kernel(const float* __restrict__ XL,
                               const float* __restrict__ ELOG,
                               const float* __restrict__ SUMB,
                               const int* __restrict__ ei,
                               float* __restrict__ YBUF, int Ep, int E) {
  int e = blockIdx.x * blockDim.x + threadIdx.x;
  if (e >= Ep) return;
  const int p = blockIdx.y;
  int src, dst;
  if (e < E) { src = ei[e]; dst = ei[E + e]; }
  else       { src = dst = e - E; }

#pragma unroll
  for (int hh = 0; hh < HH; ++hh) {
    const int ph    = p * HH + hh;
    const float alp = ELOG[(size_t)e * PH + ph] / SUMB[(size_t)dst * PH + ph];
    const float4* xv = (const float4*)(XL + (size_t)src * DD + ph * 8);
    float4 a0 = xv[0], a1 = xv[1];
    float* yb = YBUF + (size_t)dst * DD + ph * 8;
    atomicAdd(yb + 0, a0.x * alp);
    atomicAdd(yb + 1, a0.y * alp);
    atomicAdd(yb + 2, a0.z * alp);
    atomicAdd(yb + 3, a0.w * alp);
    atomicAdd(yb + 4, a1.x * alp);
    atomicAdd(yb + 5, a1.y * alp);
    atomicAdd(yb + 6, a1.z * alp);
    atomicAdd(yb + 7, a1.w * alp);
  }
}

// -------------------------- epilogue: bias + ReLU + LayerNorm ---------------
// One wave32 per node, 3 channels per lane; shuffle reductions for mean/var.
__global__ void epilogue_kernel(const float* __restrict__ YBUF,
                                const float* __restrict__ bias,
                                const float* __restrict__ ln_g,
                                const float* __restrict__ ln_b,
                                float* __restrict__ Hout, int layer, int Nn) {
  const int node = blockIdx.x * 8 + (threadIdx.x >> 5);
  const int lane = threadIdx.x & 31;
  if (node >= Nn) return;

  float v[3];
  float s = 0.0f, ss = 0.0f;
#pragma unroll
  for (int j = 0; j < 3; ++j) {
    int c = lane * 3 + j;                    // 0..95
    float y = YBUF[(size_t)node * DD + c] + bias[(size_t)layer * DD + c];
    y = fmaxf(y, 0.0f);                      // ReLU
    v[j] = y; s += y; ss += y * y;
  }
#pragma unroll
  for (int off = 16; off > 0; off >>= 1) {
    s  += __shfl_xor(s,  off, 32);
    ss += __shfl_xor(ss, off, 32);
  }
  const float mu  = s * (1.0f / DD);
  const float var = ss * (1.0f / DD) - mu * mu;
  const float inv = rsqrtf(var + LN_EPS);
#pragma unroll
  for (int j = 0; j < 3; ++j) {
    int c = lane * 3 + j;
    Hout[(size_t)node * DD + c] =
        (v[j] - mu) * inv * ln_g[(size_t)layer * DD + c] +
        ln_b[(size_t)layer * DD + c];
  }
}

// -------------------------- host driver ------------------------------------

extern "C" void kernel_launch(void* const* d_in, const int* in_sizes, int n_in,
                              void* d_out, int out_size, void* d_ws, size_t ws_size,
                              hipStream_t stream) {
  const float* x    = (const float*)d_in[0];
  const int*   ei   = (const int*)  d_in[1];
  const float* Wl   = (const float*)d_in[2];
  const float* Wr   = (const float*)d_in[3];
  const float* bl   = (const float*)d_in[4];
  const float* br   = (const float*)d_in[5];
  const float* att  = (const float*)d_in[6];
  const float* bias = (const float*)d_in[7];
  const float* ln_g = (const float*)d_in[8];
  const float* ln_b = (const float*)d_in[9];
  float* out = (float*)d_out;

  const int Nn = in_sizes[0] / DD;   // 50000
  const int E  = in_sizes[1] / 2;    // 800000
  const int Ep = E + Nn;             // edges + self loops

  // workspace layout (floats)
  float* ws   = (float*)d_ws;
  float* XL   = ws;                           // N*96
  float* XR   = XL   + (size_t)Nn * DD;       // N*96
  float* MAXB = XR   + (size_t)Nn * DD;       // N*12
  float* SUMB = MAXB + (size_t)Nn * PH;       // N*12
  float* ELOG = SUMB + (size_t)Nn * PH;       // Ep*12
  float* YBUF = ELOG + (size_t)Ep * PH;       // N*96
  float* HBUF = YBUF + (size_t)Nn * DD;       // N*96 (inter-layer features)

  const int n12 = Nn * PH;
  const dim3 blk256(256);
  const dim3 gInit((n12 + 255) / 256);
  const dim3 gGemm((Nn + 15) / 16);
  const dim3 gEdge((Ep + 255) / 256);
  const dim3 gScat((Ep + 255) / 256, PP);
  const dim3 gEpi((Nn + 7) / 8);

  for (int l = 0; l < 3; ++l) {
    const float* hin  = (l == 0) ? x   : HBUF;
    float*       hout = (l == 2) ? out : HBUF;

    init_kernel<<<gInit, blk256, 0, stream>>>(MAXB, SUMB, YBUF, n12);
    gemm_xlxr_kernel<<<gGemm, dim3(128), 0, stream>>>(hin, Wl, Wr, bl, br,
                                                      XL, XR, l, Nn);
    logits_kernel<<<gEdge, blk256, 0, stream>>>(XL, XR, ei, att, ELOG, MAXB,
                                                l, Ep, E);
    expsum_kernel<<<gEdge, blk256, 0, stream>>>(ELOG, MAXB, SUMB, ei, Ep, E);
    scatter_kernel<<<gScat, blk256, 0, stream>>>(XL, ELOG, SUMB, ei, YBUF,
                                                 Ep, E);
    epilogue_kernel<<<gEpi, blk256, 0, stream>>>(YBUF, bias, ln_g, ln_b,
                                                 hout, l, Nn);
  }
}